// Window_89696097010079
// MI455X (gfx1250) — compile-verified
//
#include <hip/hip_runtime.h>
#include <math.h>

#define B_  256
#define T_  2048
#define V_  128
#define C_  512
#define K_  10
#define N3  30

#define OUT_W     0
#define OUT_KAPPA (B_ * V_)            /* 32768 */
#define OUT_PHI   (B_ * V_ + B_ * K_)  /* 35328 */

typedef float v2f __attribute__((ext_vector_type(2)));
typedef float v4f __attribute__((ext_vector_type(4)));
typedef float v8f __attribute__((ext_vector_type(8)));

// ---------------------------------------------------------------------------
// Kernel 1: params = exp(x @ W^T + b); kappa = kappa_old + pre_kappa
// GEMM M=256, N=30 (padded to 32), K=512 via V_WMMA_F32_16X16X4_F32.
// One wave per (mtile, ntile): 16 x 2 = 32 waves, 128 WMMA k-steps each.
// ---------------------------------------------------------------------------
__global__ __launch_bounds__(32)
void k1_params_wmma(const float* __restrict__ x,        // (B, C)
                    const float* __restrict__ W,        // (3K, C)
                    const float* __restrict__ bias,     // (3K,)
                    const float* __restrict__ kappa_old,// (B, K)
                    float* __restrict__ alpha,          // ws (B, K)
                    float* __restrict__ beta,           // ws (B, K)
                    float* __restrict__ kappa,          // ws (B, K)
                    float* __restrict__ out)            // d_out
{
    const int mtile = blockIdx.x >> 1;      // 0..15
    const int ntile = blockIdx.x & 1;       // 0..1
    const int lane  = threadIdx.x;          // 0..31
    const int half  = lane >> 4;            // 0 or 1
    const int l16   = lane & 15;

    const int m = mtile * 16 + l16;         // A row held by this lane (both halves)
    const int n = ntile * 16 + l16;         // B column held by this lane
    const bool nvalid = (n < N3);

    // A 16x4 f32 layout: VGPR0/1 = K{0,1} for lanes 0-15, K{2,3} for lanes 16-31.
    const float* xrow = x + (size_t)m * C_ + 2 * half;
    // B[k][n] = W[n][k] (W stored (out,in) row-major) -> contiguous in k.
    const float* wrow = W + (size_t)(nvalid ? n : 0) * C_ + 2 * half;

    v8f acc = {};
    for (int k0 = 0; k0 < C_; k0 += 4) {
        v2f a  = *(const v2f*)(xrow + k0);
        v2f bm = *(const v2f*)(wrow + k0);
        if (!nvalid) { bm.x = 0.0f; bm.y = 0.0f; }   // pad N=30 -> 32 with zeros
        // D = A*B + C  (EXEC all-1s: full 32-thread block)
        acc = __builtin_amdgcn_wmma_f32_16x16x4_f32(
            /*neg_a=*/false, a, /*neg_b=*/false, bm,
            /*c_mod=*/(short)0, acc, /*reuse_a=*/false, /*reuse_b=*/false);
    }

    if (nvalid) {
        const float bn = bias[n];
#pragma unroll
        for (int r = 0; r < 8; ++r) {
            // C/D layout: VGPR r holds M = r (lanes 0-15) / r+8 (lanes 16-31), N = lane%16
            const int row = mtile * 16 + r + 8 * half;
            const float val = expf(acc[r] + bn);
            if (n < K_) {
                alpha[row * K_ + n] = val;
            } else if (n < 2 * K_) {
                beta[row * K_ + (n - K_)] = val;
            } else {
                const int kk  = n - 2 * K_;
                const float kp = kappa_old[row * K_ + kk] + val;
                kappa[row * K_ + kk] = kp;
                out[OUT_KAPPA + row * K_ + kk] = kp;
            }
        }
    }
}

// ---------------------------------------------------------------------------
// Kernel 2: phi[b,t] = sum_k alpha_k * exp(-beta_k * (kappa_k - t)^2)
// ---------------------------------------------------------------------------
__global__ __launch_bounds__(256)
void k2_phi(const float* __restrict__ alpha,
            const float* __restrict__ beta,
            const float* __restrict__ kappa,
            float* __restrict__ out)
{
    const int b = blockIdx.y;
    const int t = blockIdx.x * 256 + threadIdx.x;

    const float* aB = alpha + b * K_;
    const float* bB = beta  + b * K_;
    const float* kB = kappa + b * K_;

    float a[K_], be[K_], kp[K_];
#pragma unroll
    for (int k = 0; k < K_; ++k) { a[k] = aB[k]; be[k] = bB[k]; kp[k] = kB[k]; }

    const float tf = (float)t;
    float phi = 0.0f;
#pragma unroll
    for (int k = 0; k < K_; ++k) {
        const float d = kp[k] - tf;
        phi += a[k] * __expf(-be[k] * d * d);
    }
    out[OUT_PHI + (size_t)b * T_ + t] = phi;
}

// ---------------------------------------------------------------------------
// Kernel 3: the 256 MB stream.  partial[b, chunk, v] = sum over 256 t's of
// phi[b,t] * onehots[b,t,v].  Block = 8 waves; each wave owns one t per
// iteration (32 lanes x float4 NT loads = 512 B/row), LDS reduce across waves.
// Fixed summation order -> bitwise deterministic.
// ---------------------------------------------------------------------------
__global__ __launch_bounds__(256)
void k3_wpart(const float* __restrict__ onehots,   // (B, T, V)
              const float* __restrict__ phi,       // d_out phi region, (B, T)
              float* __restrict__ partial)         // ws (B, 8, V)
{
    const int b     = blockIdx.y;
    const int chunk = blockIdx.x;          // 0..7 -> t in [chunk*256, chunk*256+256)
    const int tid   = threadIdx.x;
    const int wave  = tid >> 5;            // 0..7
    const int lane  = tid & 31;            // v = lane*4 .. lane*4+3

    const int tbase = chunk * 256;
    const float* phiB = phi     + (size_t)b * T_;
    const float* oneB = onehots + (size_t)b * T_ * V_;

    v4f acc = {};
    for (int i = 0; i < 32; ++i) {
        const int t = tbase + wave + i * 8;
        const float p = phiB[t];
        const float* row = oneB + (size_t)t * V_;
        v4f o = __builtin_nontemporal_load((const v4f*)row + lane);
        if (i + 1 < 32)
            __builtin_prefetch(oneB + (size_t)(t + 8) * V_ + lane * 4, 0, 0);
        acc += p * o;   // read-once data: NT keeps it out of the way of L2
    }

    __shared__ float red[8][V_];
    red[wave][lane * 4 + 0] = acc.x;
    red[wave][lane * 4 + 1] = acc.y;
    red[wave][lane * 4 + 2] = acc.z;
    red[wave][lane * 4 + 3] = acc.w;
    __syncthreads();

    if (tid < V_) {
        float s = 0.0f;
#pragma unroll
        for (int w = 0; w < 8; ++w) s += red[w][tid];
        partial[((size_t)b * 8 + chunk) * V_ + tid] = s;
    }
}

// ---------------------------------------------------------------------------
// Kernel 4: w[b,v] = sum_chunk partial[b, chunk, v]  (fixed order)
// ---------------------------------------------------------------------------
__global__ __launch_bounds__(256)
void k4_wreduce(const float* __restrict__ partial, float* __restrict__ out)
{
    const int idx = blockIdx.x * 256 + threadIdx.x;   // < B*V, idx = b*V + v
    const int b = idx >> 7;
    const int v = idx & (V_ - 1);
    float s = 0.0f;
#pragma unroll
    for (int c = 0; c < 8; ++c) s += partial[((size_t)b * 8 + c) * V_ + v];
    out[OUT_W + idx] = s;
}

// ---------------------------------------------------------------------------
extern "C" void kernel_launch(void* const* d_in, const int* in_sizes, int n_in,
                              void* d_out, int out_size, void* d_ws, size_t ws_size,
                              hipStream_t stream) {
    const float* x         = (const float*)d_in[0];   // (B, C)
    const float* kappa_old = (const float*)d_in[1];   // (B, K)
    const float* onehots   = (const float*)d_in[2];   // (B, T, V)
    const float* W         = (const float*)d_in[3];   // (3K, C)
    const float* bias      = (const float*)d_in[4];   // (3K,)

    float* out = (float*)d_out;
    float* ws  = (float*)d_ws;
    float* alpha   = ws;                     // B*K
    float* beta    = ws + 1 * B_ * K_;       // B*K
    float* kappa   = ws + 2 * B_ * K_;       // B*K
    float* partial = ws + 3 * B_ * K_;       // B*8*V

    k1_params_wmma<<<32, 32, 0, stream>>>(x, W, bias, kappa_old,
                                          alpha, beta, kappa, out);
    k2_phi<<<dim3(T_ / 256, B_), 256, 0, stream>>>(alpha, beta, kappa, out);
    k3_wpart<<<dim3(8, B_), 256, 0, stream>>>(onehots, out + OUT_PHI, partial);
    k4_wreduce<<<(B_ * V_) / 256, 256, 0, stream>>>(partial, out);
}